// neuroheed_50319836840342
// MI455X (gfx1250) — compile-verified
//
#include <hip/hip_runtime.h>

typedef __attribute__((ext_vector_type(2))) float v2f;
typedef __attribute__((ext_vector_type(8))) float v8f;

#define MBATCH 8
#define TSAMP 64000
#define NENC 256
#define LWIN 20
#define BCH 64
#define KCH 100
#define SCH 130
#define DFRM 6399
#define DEEG 64
#define TEEG 512
#define NST 64

// ---------------------------------------------------------------------------
// Generic fully-strided batched GEMM on fp32 WMMA (16x16x4).
// C[b][m,n] = act( sum_k A[b][m,k]*B[b][k,n] + bias )
// Each wave computes a 16(M) x 32(N) tile: two accumulators share one A frag.
// Fast path (full tile) does unguarded loads; edge path clamps addresses and
// masks values (branch-free, no EXEC manipulation in the inner loop).
// ---------------------------------------------------------------------------
struct Gemm {
  const float* A; const float* B; float* C; const float* bias;
  long sAm, sAk, sAb;
  long sBk, sBn, sBb;
  long sCm, sCn, sCb;
  int M, N, K, batch, biasMode, act; // biasMode: 0 none, 1 per-col, 2 per-row; act: 0 none, 1 relu
};

__global__ __launch_bounds__(128) void gemm_wmma_f32(Gemm g) {
  const int wave = threadIdx.x >> 5;              // uniform per wave
  const int lane = threadIdx.x & 31;
  const int nBase = (blockIdx.y * 8 + wave * 2) * 16;  // wave covers cols [nBase, nBase+32)
  if (nBase >= g.N) return;                       // wave-uniform exit
  const int m0 = blockIdx.x * 16;
  const long bz = blockIdx.z;
  const float* A = g.A + bz * g.sAb;
  const float* B = g.B + bz * g.sBb;
  float*       C = g.C + bz * g.sCb;
  const int hi = lane >> 4;                       // 0 or 1 (lane half)
  const int lo = lane & 15;
  const int mA  = m0 + lo;
  const int nB0 = nBase + lo;
  const int nB1 = nBase + 16 + lo;

  const long sAk = g.sAk, sBk = g.sBk;
  const long dA = 4 * sAk, dB = 4 * sBk;

  v8f acc0 = {0.f,0.f,0.f,0.f,0.f,0.f,0.f,0.f};
  v8f acc1 = {0.f,0.f,0.f,0.f,0.f,0.f,0.f,0.f};
  const int kMain = g.K & ~3;
  const bool full = (m0 + 16 <= g.M) && (nBase + 32 <= g.N);  // wave-uniform

  if (full) {
    // -------- fast path: no guards, pointer-bumped loads, 2 wmma per step ---
    const float* pA  = A + (long)mA * g.sAm + (long)(2 * hi) * sAk;
    const float* pB0 = B + (long)nB0 * g.sBn + (long)(2 * hi) * sBk;
    const float* pB1 = B + (long)nB1 * g.sBn + (long)(2 * hi) * sBk;
    for (int k0 = 0; k0 < kMain; k0 += 4) {
      v2f av, b0, b1;
      av.x = pA[0];        av.y = pA[sAk];
      b0.x = pB0[0];       b0.y = pB0[sBk];
      b1.x = pB1[0];       b1.y = pB1[sBk];
      __builtin_prefetch(pA + 2 * dA, 0, 1);      // global_prefetch_b8
      __builtin_prefetch(pB0 + 2 * dB, 0, 1);
      acc0 = __builtin_amdgcn_wmma_f32_16x16x4_f32(false, av, false, b0, (short)0, acc0, false, false);
      acc1 = __builtin_amdgcn_wmma_f32_16x16x4_f32(false, av, false, b1, (short)0, acc1, false, false);
      pA += dA; pB0 += dB; pB1 += dB;
    }
    if (kMain < g.K) {                            // K tail (K % 4 != 0): clamp + mask
      const int ka = kMain + 2 * hi;
      const int kb = ka + 1;
      const float mk0 = (ka < g.K) ? 1.f : 0.f;
      const float mk1 = (kb < g.K) ? 1.f : 0.f;
      const long kac = (ka < g.K) ? ka : (g.K - 1);
      const long kbc = (kb < g.K) ? kb : (g.K - 1);
      v2f av, b0, b1;
      av.x = mk0 * A[(long)mA * g.sAm + kac * sAk];
      av.y = mk1 * A[(long)mA * g.sAm + kbc * sAk];
      b0.x = mk0 * B[kac * sBk + (long)nB0 * g.sBn];
      b0.y = mk1 * B[kbc * sBk + (long)nB0 * g.sBn];
      b1.x = mk0 * B[kac * sBk + (long)nB1 * g.sBn];
      b1.y = mk1 * B[kbc * sBk + (long)nB1 * g.sBn];
      acc0 = __builtin_amdgcn_wmma_f32_16x16x4_f32(false, av, false, b0, (short)0, acc0, false, false);
      acc1 = __builtin_amdgcn_wmma_f32_16x16x4_f32(false, av, false, b1, (short)0, acc1, false, false);
    }
#pragma unroll
    for (int i = 0; i < 8; ++i) {
      const int m = m0 + i + 8 * hi;
      float v0 = acc0[i], v1 = acc1[i];
      if (g.biasMode == 1) { v0 += g.bias[nB0]; v1 += g.bias[nB1]; }
      else if (g.biasMode == 2) { float bm = g.bias[m]; v0 += bm; v1 += bm; }
      if (g.act == 1) { v0 = fmaxf(v0, 0.f); v1 = fmaxf(v1, 0.f); }
      C[(long)m * g.sCm + (long)nB0 * g.sCn] = v0;
      C[(long)m * g.sCm + (long)nB1 * g.sCn] = v1;
    }
  } else {
    // -------- edge path: clamp addresses, mask values (branch-free) --------
    const int mC  = (mA  < g.M) ? mA  : (g.M - 1);
    const int n0C = (nB0 < g.N) ? nB0 : (g.N - 1);
    const int n1C = (nB1 < g.N) ? nB1 : (g.N - 1);
    const float fA = (mA  < g.M) ? 1.f : 0.f;
    const float f0 = (nB0 < g.N) ? 1.f : 0.f;
    const float f1 = (nB1 < g.N) ? 1.f : 0.f;
    const float* pA  = A + (long)mC * g.sAm + (long)(2 * hi) * sAk;
    const float* pB0 = B + (long)n0C * g.sBn + (long)(2 * hi) * sBk;
    const float* pB1 = B + (long)n1C * g.sBn + (long)(2 * hi) * sBk;
    for (int k0 = 0; k0 < kMain; k0 += 4) {
      v2f av, b0, b1;
      av.x = fA * pA[0];    av.y = fA * pA[sAk];
      b0.x = f0 * pB0[0];   b0.y = f0 * pB0[sBk];
      b1.x = f1 * pB1[0];   b1.y = f1 * pB1[sBk];
      acc0 = __builtin_amdgcn_wmma_f32_16x16x4_f32(false, av, false, b0, (short)0, acc0, false, false);
      acc1 = __builtin_amdgcn_wmma_f32_16x16x4_f32(false, av, false, b1, (short)0, acc1, false, false);
      pA += dA; pB0 += dB; pB1 += dB;
    }
    if (kMain < g.K) {                            // K tail: clamp k too
      const int ka = kMain + 2 * hi;
      const int kb = ka + 1;
      const float mk0 = fA * ((ka < g.K) ? 1.f : 0.f);
      const float mk1 = fA * ((kb < g.K) ? 1.f : 0.f);
      const float nk0 = (ka < g.K) ? 1.f : 0.f;
      const float nk1 = (kb < g.K) ? 1.f : 0.f;
      const long kac = (ka < g.K) ? ka : (g.K - 1);
      const long kbc = (kb < g.K) ? kb : (g.K - 1);
      v2f av, b0, b1;
      av.x = mk0 * A[(long)mC * g.sAm + kac * sAk];
      av.y = mk1 * A[(long)mC * g.sAm + kbc * sAk];
      b0.x = f0 * nk0 * B[kac * sBk + (long)n0C * g.sBn];
      b0.y = f0 * nk1 * B[kbc * sBk + (long)n0C * g.sBn];
      b1.x = f1 * nk0 * B[kac * sBk + (long)n1C * g.sBn];
      b1.y = f1 * nk1 * B[kbc * sBk + (long)n1C * g.sBn];
      acc0 = __builtin_amdgcn_wmma_f32_16x16x4_f32(false, av, false, b0, (short)0, acc0, false, false);
      acc1 = __builtin_amdgcn_wmma_f32_16x16x4_f32(false, av, false, b1, (short)0, acc1, false, false);
    }
#pragma unroll
    for (int i = 0; i < 8; ++i) {
      const int m = m0 + i + 8 * hi;
      if (m >= g.M) continue;
      float v0 = acc0[i], v1 = acc1[i];
      const bool n0OK = (nB0 < g.N), n1OK = (nB1 < g.N);
      if (g.biasMode == 1) { if (n0OK) v0 += g.bias[nB0]; if (n1OK) v1 += g.bias[nB1]; }
      else if (g.biasMode == 2) { float bm = g.bias[m]; v0 += bm; v1 += bm; }
      if (g.act == 1) { v0 = fmaxf(v0, 0.f); v1 = fmaxf(v1, 0.f); }
      if (n0OK) C[(long)m * g.sCm + (long)nB0 * g.sCn] = v0;
      if (n1OK) C[(long)m * g.sCm + (long)nB1 * g.sCn] = v1;
    }
  }
}

static void gemm(hipStream_t st,
                 const float* A, long sAm, long sAk, long sAb,
                 const float* B, long sBk, long sBn, long sBb,
                 float* C, long sCm, long sCn, long sCb,
                 const float* bias, int biasMode, int act,
                 int M, int N, int K, int batch) {
  Gemm g{A, B, C, bias, sAm, sAk, sAb, sBk, sBn, sBb, sCm, sCn, sCb, M, N, K, batch, biasMode, act};
  dim3 grid((unsigned)((M + 15) / 16), (unsigned)((N + 127) / 128), (unsigned)batch);
  gemm_wmma_f32<<<grid, 128, 0, st>>>(g);
}

// ---------------------------------------------------------------------------
// Elementwise / reduction kernels
// ---------------------------------------------------------------------------
__global__ void k_enc_conv(const float* mix, const float* w, float* mw, int total) {
  int i = blockIdx.x * blockDim.x + threadIdx.x;
  if (i >= total) return;
  int d = i % DFRM;
  int n = (i / DFRM) % NENC;
  int m = i / (DFRM * NENC);
  const float* mp = mix + (long)m * TSAMP + d * 10;
  const float* wp = w + n * LWIN;
  float s = 0.f;
#pragma unroll
  for (int l = 0; l < LWIN; ++l) s += mp[l] * wp[l];
  mw[i] = fmaxf(s, 0.f);
}

__global__ void k_gn_reduce(const float* x, float* stats, long chunk, float eps) {
  int b = blockIdx.x;
  const float* p = x + (long)b * chunk;
  float s = 0.f, ss = 0.f;
  for (long i = threadIdx.x; i < chunk; i += blockDim.x) { float v = p[i]; s += v; ss += v * v; }
  __shared__ float sh1[256], sh2[256];
  sh1[threadIdx.x] = s; sh2[threadIdx.x] = ss; __syncthreads();
  for (int st = 128; st > 0; st >>= 1) {
    if ((int)threadIdx.x < st) { sh1[threadIdx.x] += sh1[threadIdx.x + st]; sh2[threadIdx.x] += sh2[threadIdx.x + st]; }
    __syncthreads();
  }
  if (threadIdx.x == 0) {
    float m = sh1[0] / (float)chunk;
    float v = sh2[0] / (float)chunk - m * m;
    stats[2 * b] = m;
    stats[2 * b + 1] = rsqrtf(v + eps);
  }
}

__global__ void k_gn_apply_c(const float* x, const float* stats, const float* w, const float* bb,
                             float* out, int D, int C, int total) {
  int i = blockIdx.x * blockDim.x + threadIdx.x;
  if (i >= total) return;
  int n = (i / D) % C;
  int b = i / (D * C);
  out[i] = (x[i] - stats[2 * b]) * stats[2 * b + 1] * w[n] + bb[n];
}

__global__ void k_posenc(const float* eeg, float* e, int total) {
  int i = blockIdx.x * blockDim.x + threadIdx.x;
  if (i >= total) return;
  int c = i % DEEG;
  int m = (i / DEEG) % MBATCH;
  int t = i / (DEEG * MBATCH);
  int j = c >> 1;
  float f = expf(-(float)(2 * j) * (9.210340371976184f / 64.f)); // ln(10000)/Deeg
  float ang = (float)t * f;
  float pe = (c & 1) ? cosf(ang) : sinf(ang);
  e[i] = eeg[((long)m * TEEG + t) * DEEG + c] + pe;
}

__global__ void k_softmax(float* x, int cols, float scale) {
  int row = blockIdx.x;
  float* p = x + (long)row * cols;
  __shared__ float red[256];
  int tid = threadIdx.x;
  float mx = -3.0e38f;
  for (int c = tid; c < cols; c += blockDim.x) mx = fmaxf(mx, p[c] * scale);
  red[tid] = mx; __syncthreads();
  for (int s = 128; s > 0; s >>= 1) { if (tid < s) red[tid] = fmaxf(red[tid], red[tid + s]); __syncthreads(); }
  mx = red[0]; __syncthreads();
  float sum = 0.f;
  for (int c = tid; c < cols; c += blockDim.x) { float e = expf(p[c] * scale - mx); p[c] = e; sum += e; }
  red[tid] = sum; __syncthreads();
  for (int s = 128; s > 0; s >>= 1) { if (tid < s) red[tid] += red[tid + s]; __syncthreads(); }
  float inv = 1.f / red[0];
  for (int c = tid; c < cols; c += blockDim.x) p[c] *= inv;
}

__global__ void k_res_ln64(float* x, const float* r, const float* w, const float* b) {
  int row = blockIdx.x;
  int c = threadIdx.x;           // blockDim 64
  long base = (long)row * 64;
  float v = x[base + c] + r[base + c];
  __shared__ float s1[64], s2[64];
  s1[c] = v; s2[c] = v * v; __syncthreads();
  for (int st = 32; st > 0; st >>= 1) { if (c < st) { s1[c] += s1[c + st]; s2[c] += s2[c + st]; } __syncthreads(); }
  float mean = s1[0] * (1.f / 64.f);
  float var = s2[0] * (1.f / 64.f) - mean * mean;
  float rs = rsqrtf(var + 1e-5f);
  x[base + c] = (v - mean) * rs * w[c] + b[c];
}

__global__ void k_interp(const float* e, float* cat, int total) {
  int i = blockIdx.x * blockDim.x + threadIdx.x;
  if (i >= total) return;
  int d = i % DFRM;
  int c = (i / DFRM) % DEEG;
  int m = i / (DFRM * DEEG);
  float coord = ((float)d + 0.5f) * (512.f / 6399.f) - 0.5f;
  coord = fminf(fmaxf(coord, 0.f), 511.f);
  int i0 = (int)floorf(coord);
  int i1 = i0 + 1; if (i1 > 511) i1 = 511;
  float w = coord - (float)i0;
  float v = e[((long)i0 * MBATCH + m) * DEEG + c] * (1.f - w) + e[((long)i1 * MBATCH + m) * DEEG + c] * w;
  cat[((long)m * 128 + 64 + c) * DFRM + d] = v;
}

__global__ void k_segment(const float* xf, float* seg, int total) {
  int i = blockIdx.x * blockDim.x + threadIdx.x;
  if (i >= total) return;
  int s = i % SCH;
  int k = (i / SCH) % KCH;
  int n = (i / (SCH * KCH)) % BCH;
  int b = i / (SCH * KCH * BCH);
  int pos = (s >> 1) * 100 + (s & 1) * 50 + k - 50;
  seg[i] = (pos >= 0 && pos < DFRM) ? xf[((long)(b * BCH + n)) * DFRM + pos] : 0.f;
}

__global__ void k_tr_intra(const float* seg, float* u, int total) {
  int i = blockIdx.x * blockDim.x + threadIdx.x;
  if (i >= total) return;
  int k = i % KCH;
  int n = (i / KCH) % BCH;
  int bs = i / (KCH * BCH);
  int b = bs / SCH, s = bs % SCH;
  u[i] = seg[(((long)(b * BCH + n)) * KCH + k) * SCH + s];
}

__global__ void k_tr_inter(const float* x4, float* u, int total) {
  int i = blockIdx.x * blockDim.x + threadIdx.x;
  if (i >= total) return;
  int s = i % SCH;
  int n = (i / SCH) % BCH;
  int bk = i / (SCH * BCH);
  int b = bk / KCH, k = bk % KCH;
  u[i] = x4[(((long)(b * BCH + n)) * KCH + k) * SCH + s];
}

__global__ void k_s4_kern(const float* log_dt, const float* A_logre, const float* A_im,
                          const float* C_re, const float* C_im, float* kern, int L) {
  int i = blockIdx.x * blockDim.x + threadIdx.x;
  if (i >= 64 * L) return;
  int h = i / L, l = i % L;
  float dt = expf(log_dt[h]);
  float fl = (float)l;
  float acc = 0.f;
  for (int n = 0; n < NST; ++n) {
    float Are = -expf(A_logre[h * NST + n]);
    float Aim = A_im[h * NST + n];
    float dRe = dt * Are, dIm = dt * Aim;
    float ex = expf(dRe);
    float dA_re = ex * cosf(dIm), dA_im = ex * sinf(dIm);
    float den = Are * Are + Aim * Aim;
    float nre = dA_re - 1.f, nim = dA_im;
    float dB_re = (nre * Are + nim * Aim) / den;
    float dB_im = (nim * Are - nre * Aim) / den;
    float cre = C_re[h * NST + n], cim = C_im[h * NST + n];
    float Kre = cre * dB_re - cim * dB_im;
    float Kim = cre * dB_im + cim * dB_re;
    float e2 = expf(dRe * fl);
    float vr = e2 * cosf(dIm * fl), vi = e2 * sinf(dIm * fl);
    acc += Kre * vr - Kim * vi;
  }
  kern[i] = 2.f * acc;
}

__global__ void k_toep(const float* kern, float* toep, int L, int total) {
  int i = blockIdx.x * blockDim.x + threadIdx.x;
  if (i >= total) return;
  int t = i % L;
  int s = (i / L) % L;
  int h = i / (L * L);
  toep[i] = (t >= s) ? kern[h * L + (t - s)] : 0.f;
}

__global__ void k_gelu_d(float* y, const float* u, const float* D, int L, int total) {
  int i = blockIdx.x * blockDim.x + threadIdx.x;
  if (i >= total) return;
  int h = (i / L) % BCH;
  float x = y[i] + u[i] * D[h];
  float t = tanhf(0.7978845608028654f * (x + 0.044715f * x * x * x));
  y[i] = 0.5f * x * (1.f + t);
}

__global__ void k_glu(const float* y2, float* out, int L, int total) {
  int i = blockIdx.x * blockDim.x + threadIdx.x;
  if (i >= total) return;
  int hl = i % (BCH * L);
  int b = i / (BCH * L);
  float a = y2[(long)b * (128 * L) + hl];
  float g = y2[(long)b * (128 * L) + 64 * L + hl];
  out[i] = a * (1.f / (1.f + expf(-g)));
}

__global__ void k_gn_apply_intra(const float* lin, const float* stats, const float* w, const float* bb,
                                 const float* seg, float* out, int total) {
  int j = blockIdx.x * blockDim.x + threadIdx.x;
  if (j >= total) return;
  int s = j % SCH;
  int k = (j / SCH) % KCH;
  int n = (j / (SCH * KCH)) % BCH;
  int b = j / (SCH * KCH * BCH);
  long li = ((long)(b * SCH + s)) * (BCH * KCH) + n * KCH + k;
  out[j] = (lin[li] - stats[2 * b]) * stats[2 * b + 1] * w[n] + bb[n] + seg[j];
}

__global__ void k_gn_apply_inter(const float* lin, const float* stats, const float* w, const float* bb,
                                 const float* add, float* out, int total) {
  int j = blockIdx.x * blockDim.x + threadIdx.x;
  if (j >= total) return;
  int s = j % SCH;
  int k = (j / SCH) % KCH;
  int n = (j / (SCH * KCH)) % BCH;
  int b = j / (SCH * KCH * BCH);
  long li = ((long)(b * KCH + k)) * (BCH * SCH) + n * SCH + s;
  out[j] = (lin[li] - stats[2 * b]) * stats[2 * b + 1] * w[n] + bb[n] + add[j];
}

__global__ void k_overadd_prelu(const float* seg, const float* prelu, float* out, int total) {
  int i = blockIdx.x * blockDim.x + threadIdx.x;
  if (i >= total) return;
  int p = i % DFRM;
  int n = (i / DFRM) % BCH;
  int b = i / (DFRM * BCH);
  long base = ((long)(b * BCH + n)) * KCH;
  int q = p + 50;
  int s1 = 2 * (q / 100), k1 = q % 100;
  int s2 = 2 * (p / 100) + 1, k2 = p % 100;
  float v = seg[(base + k1) * SCH + s1] + seg[(base + k2) * SCH + s2];
  float a = prelu[0];
  out[i] = (v >= 0.f) ? v : a * v;
}

__global__ void k_mul_inplace(float* a, const float* b, int total) {
  int i = blockIdx.x * blockDim.x + threadIdx.x;
  if (i >= total) return;
  a[i] *= b[i];
}

__global__ void k_overlap(const float* est2, float* out, int total) {
  int i = blockIdx.x * blockDim.x + threadIdx.x;
  if (i >= total) return;
  int t = i % TSAMP;
  int m = i / TSAMP;
  int d = t / 10, j = t % 10;
  float v = 0.f;
  if (d < DFRM) v += est2[((long)m * DFRM + d) * 20 + j];
  if (d >= 1)   v += est2[((long)m * DFRM + d - 1) * 20 + j + 10];
  out[i] = v;
}

// ---------------------------------------------------------------------------
// Host driver
// ---------------------------------------------------------------------------
extern "C" void kernel_launch(void* const* d_in, const int* in_sizes, int n_in,
                              void* d_out, int out_size, void* d_ws, size_t ws_size,
                              hipStream_t stream) {
  (void)in_sizes; (void)n_in; (void)out_size; (void)ws_size;
  const float* mixture = (const float*)d_in[0];
  const float* eegin   = (const float*)d_in[1];
  auto P = [&](int i) { return (const float*)d_in[2 + i]; };

  // jax tree order: blocks (6 x 24 leaves), then sorted top-level, then tf leaves.
  const float* bott_w  = P(144);
  const float* dec_w   = P(145);
  const float* enc_w   = P(146);
  const float* fus_w   = P(147);
  const float* ln_b    = P(148);
  const float* ln_w    = P(149);
  const float* mask_w  = P(150);
  const float* prelu_a = P(151);
  const float* tf_ff1_b = P(152);
  const float* tf_ff1_w = P(153);
  const float* tf_ff2_b = P(154);
  const float* tf_ff2_w = P(155);
  const float* tf_in_b  = P(156);
  const float* tf_in_w  = P(157);
  const float* tf_ln1_b = P(158);
  const float* tf_ln1_w = P(159);
  const float* tf_ln2_b = P(160);
  const float* tf_ln2_w = P(161);
  const float* tf_out_b = P(162);
  const float* tf_out_w = P(163);

  // workspace layout (floats)
  float* ws = (float*)d_ws;
  size_t off = 0;
  auto take = [&](size_t n) -> float* { float* p = ws + off; off += (n + 63) & ~(size_t)63; return p; };
  const int TMW = MBATCH * NENC * DFRM;             // 13,105,152
  const int TSEG = MBATCH * BCH * KCH * SCH;        // 6,656,000
  float* mw    = take(TMW);
  float* xn    = take(TMW);        // later reused as mask & masked product
  float* cat   = take(MBATCH * 128 * DFRM);
  float* ebuf  = take(TEEG * MBATCH * DEEG);
  float* qkv   = take(TEEG * MBATCH * 192);
  float* att   = take(MBATCH * TEEG * TEEG);
  float* obuf  = take(TEEG * MBATCH * DEEG);
  float* obuf2 = take(TEEG * MBATCH * DEEG);
  float* hbuf  = take(TEEG * MBATCH * 256);
  float* xf    = take(MBATCH * BCH * DFRM);
  float* seg   = take(TSEG);
  float* ubuf  = take(TSEG);
  float* ybuf  = take(TSEG);
  float* y2buf = take(2 * TSEG);
  float* x4    = take(TSEG);
  float* kern  = take(64 * SCH);
  float* toep  = take(64 * SCH * SCH);
  float* est2  = take(MBATCH * DFRM * 20);
  float* stats = take(64);

  auto gs = [](int total) { return dim3((unsigned)((total + 255) / 256)); };

  // 1) encoder conv + relu
  k_enc_conv<<<gs(TMW), 256, 0, stream>>>(mixture, enc_w, mw, TMW);
  // 2) group norm over (C,D) per sample
  k_gn_reduce<<<MBATCH, 256, 0, stream>>>(mw, stats, (long)NENC * DFRM, 1e-8f);
  k_gn_apply_c<<<gs(TMW), 256, 0, stream>>>(mw, stats, ln_w, ln_b, xn, DFRM, NENC, TMW);
  // 3) bottleneck -> cat rows [0,64)
  gemm(stream, bott_w, 256, 1, 0, xn, DFRM, 1, (long)NENC * DFRM,
       cat, DFRM, 1, (long)128 * DFRM, nullptr, 0, 0, 64, DFRM, 256, MBATCH);
  // 4) EEG transformer
  k_posenc<<<gs(TEEG * MBATCH * DEEG), 256, 0, stream>>>(eegin, ebuf, TEEG * MBATCH * DEEG);
  for (int i = 0; i < 5; ++i) {
    const float* in_w  = tf_in_w  + (long)i * 192 * 64;
    const float* in_b  = tf_in_b  + i * 192;
    const float* out_w = tf_out_w + (long)i * 64 * 64;
    const float* out_b = tf_out_b + i * 64;
    const float* ln1w = tf_ln1_w + i * 64;
    const float* ln1b = tf_ln1_b + i * 64;
    const float* ln2w = tf_ln2_w + i * 64;
    const float* ln2b = tf_ln2_b + i * 64;
    const float* ff1w = tf_ff1_w + (long)i * 256 * 64;
    const float* ff1b = tf_ff1_b + i * 256;
    const float* ff2w = tf_ff2_w + (long)i * 64 * 256;
    const float* ff2b = tf_ff2_b + i * 64;
    // qkv[tm,f] = e[tm,:]*in_w[f,:]^T + in_b
    gemm(stream, ebuf, 64, 1, 0, in_w, 1, 64, 0, qkv, 192, 1, 0, in_b, 1, 0, 4096, 192, 64, 1);
    // att[m,t,s] = Q_m K_m^T
    gemm(stream, qkv, 1536, 1, 192, qkv + 64, 1, 1536, 192,
         att, 512, 1, (long)512 * 512, nullptr, 0, 0, 512, 512, 64, MBATCH);
    k_softmax<<<4096, 256, 0, stream>>>(att, 512, 0.125f);
    // o[t,m,e] = att V
    gemm(stream, att, 512, 1, (long)512 * 512, qkv + 128, 1536, 1, 192,
         obuf, 512, 1, 64, nullptr, 0, 0, 512, 64, 512, MBATCH);
    // out proj
    gemm(stream, obuf, 64, 1, 0, out_w, 1, 64, 0, obuf2, 64, 1, 0, out_b, 1, 0, 4096, 64, 64, 1);
    k_res_ln64<<<4096, 64, 0, stream>>>(ebuf, obuf2, ln1w, ln1b);
    // ffn
    gemm(stream, ebuf, 64, 1, 0, ff1w, 1, 64, 0, hbuf, 256, 1, 0, ff1b, 1, 1, 4096, 256, 64, 1);
    gemm(stream, hbuf, 256, 1, 0, ff2w, 1, 256, 0, obuf2, 64, 1, 0, ff2b, 1, 0, 4096, 64, 256, 1);
    k_res_ln64<<<4096, 64, 0, stream>>>(ebuf, obuf2, ln2w, ln2b);
  }
  // interp EEG to cat rows [64,128)
  k_interp<<<gs(MBATCH * DEEG * DFRM), 256, 0, stream>>>(ebuf, cat, MBATCH * DEEG * DFRM);
  // 5) fusion
  gemm(stream, fus_w, 128, 1, 0, cat, DFRM, 1, (long)128 * DFRM,
       xf, DFRM, 1, (long)BCH * DFRM, nullptr, 0, 0, 64, DFRM, 128, MBATCH);
  // 6) segmentation
  k_segment<<<gs(TSEG), 256, 0, stream>>>(xf, seg, TSEG);
  // 7) dual-path S4 blocks
  for (int r = 0; r < 6; ++r) {
    auto BP = [&](int j) { return P(r * 24 + j); };
    const float* inter_gn_b = BP(0);  const float* inter_gn_w = BP(1);
    const float* inter_lin_b = BP(2); const float* inter_lin_w = BP(3);
    const float* inter_A_im = BP(4);  const float* inter_A_logre = BP(5);
    const float* inter_C_im = BP(6);  const float* inter_C_re = BP(7);
    const float* inter_D = BP(8);     const float* inter_log_dt = BP(9);
    const float* inter_out_b = BP(10); const float* inter_out_w = BP(11);
    const float* intra_gn_b = BP(12); const float* intra_gn_w = BP(13);
    const float* intra_lin_b = BP(14); const float* intra_lin_w = BP(15);
    const float* intra_A_im = BP(16); const float* intra_A_logre = BP(17);
    const float* intra_C_im = BP(18); const float* intra_C_re = BP(19);
    const float* intra_D = BP(20);    const float* intra_log_dt = BP(21);
    const float* intra_out_b = BP(22); const float* intra_out_w = BP(23);

    // ---- intra (L = 100, batch rows = 8*130 = 1040) ----
    k_tr_intra<<<gs(TSEG), 256, 0, stream>>>(seg, ubuf, TSEG);
    k_s4_kern<<<gs(64 * KCH), 256, 0, stream>>>(intra_log_dt, intra_A_logre, intra_A_im,
                                                intra_C_re, intra_C_im, kern, KCH);
    k_toep<<<gs(64 * KCH * KCH), 256, 0, stream>>>(kern, toep, KCH, 64 * KCH * KCH);
    gemm(stream, ubuf, 64 * KCH, 1, KCH, toep, KCH, 1, (long)KCH * KCH,
         ybuf, 64 * KCH, 1, KCH, nullptr, 0, 0, 1040, KCH, KCH, 64);
    k_gelu_d<<<gs(TSEG), 256, 0, stream>>>(ybuf, ubuf, intra_D, KCH, TSEG);
    gemm(stream, intra_out_w, 64, 1, 0, ybuf, KCH, 1, (long)64 * KCH,
         y2buf, KCH, 1, (long)128 * KCH, intra_out_b, 2, 0, 128, KCH, 64, 1040);
    k_glu<<<gs(TSEG), 256, 0, stream>>>(y2buf, ubuf, KCH, TSEG);
    gemm(stream, ubuf, 64, 1, 0, intra_lin_w, 1, 64, 0,
         ybuf, 64, 1, 0, intra_lin_b, 1, 0, 104000, 64, 64, 1);
    k_gn_reduce<<<MBATCH, 256, 0, stream>>>(ybuf, stats, (long)BCH * KCH * SCH, 1e-8f);
    k_gn_apply_intra<<<gs(TSEG), 256, 0, stream>>>(ybuf, stats, intra_gn_w, intra_gn_b, seg, x4, TSEG);

    // ---- inter (L = 130, batch rows = 8*100 = 800) ----
    k_tr_inter<<<gs(TSEG), 256, 0, stream>>>(x4, ubuf, TSEG);
    k_s4_kern<<<gs(64 * SCH), 256, 0, stream>>>(inter_log_dt, inter_A_logre, inter_A_im,
                                                inter_C_re, inter_C_im, kern, SCH);
    k_toep<<<gs(64 * SCH * SCH), 256, 0, stream>>>(kern, toep, SCH, 64 * SCH * SCH);
    gemm(stream, ubuf, 64 * SCH, 1, SCH, toep, SCH, 1, (long)SCH * SCH,
         ybuf, 64 * SCH, 1, SCH, nullptr, 0, 0, 800, SCH, SCH, 64);
    k_gelu_d<<<gs(TSEG), 256, 0, stream>>>(ybuf, ubuf, inter_D, SCH, TSEG);
    gemm(stream, inter_out_w, 64, 1, 0, ybuf, SCH, 1, (long)64 * SCH,
         y2buf, SCH, 1, (long)128 * SCH, inter_out_b, 2, 0, 128, SCH, 64, 800);
    k_glu<<<gs(TSEG), 256, 0, stream>>>(y2buf, ubuf, SCH, TSEG);
    gemm(stream, ubuf, 64, 1, 0, inter_lin_w, 1, 64, 0,
         ybuf, 64, 1, 0, inter_lin_b, 1, 0, 104000, 64, 64, 1);
    k_gn_reduce<<<MBATCH, 256, 0, stream>>>(ybuf, stats, (long)BCH * KCH * SCH, 1e-8f);
    k_gn_apply_inter<<<gs(TSEG), 256, 0, stream>>>(ybuf, stats, inter_gn_w, inter_gn_b, x4, seg, TSEG);
  }
  // 8) overlap-add of chunks + PReLU
  k_overadd_prelu<<<gs(MBATCH * BCH * DFRM), 256, 0, stream>>>(seg, prelu_a, xf, MBATCH * BCH * DFRM);
  // 9) mask (relu), reuse xn buffer
  gemm(stream, mask_w, 64, 1, 0, xf, DFRM, 1, (long)BCH * DFRM,
       xn, DFRM, 1, (long)NENC * DFRM, nullptr, 0, 1, 256, DFRM, 64, MBATCH);
  // 10) est = mw * mask (in place in xn)
  k_mul_inplace<<<gs(TMW), 256, 0, stream>>>(xn, mw, TMW);
  // 11) decoder: est2[m,d,l] = sum_n est[m,n,d] * dec_w[l,n]
  gemm(stream, xn, 1, DFRM, (long)NENC * DFRM, dec_w, 1, 256, 0,
       est2, 20, 1, (long)DFRM * 20, nullptr, 0, 0, DFRM, 20, 256, MBATCH);
  // 12) overlap-and-add to output (exactly 64000 samples, pad = 0)
  k_overlap<<<gs(MBATCH * TSAMP), 256, 0, stream>>>(est2, (float*)d_out, MBATCH * TSAMP);
}